// KBPNLoss_59133109731454
// MI455X (gfx1250) — compile-verified
//
#include <hip/hip_runtime.h>
#include <hip/hip_bf16.h>

// ---------------------------------------------------------------------------
// KBPN loss for MI455X (gfx1250, wave32, WMMA).
//   k_rweights : 16-tap antialiased Keys-cubic (a=-0.5) stride-4 resize filter
//   k_ksum     : WMMA reduction of kernel_pred over 64x64 spatial (C = A*ones),
//                8-way spatial split, branch-free, 8-deep load pipelining
//   k_hrl1     : deterministic 2-stage |hr_pred - hr_target| partials (b128)
//   k_knorm    : normalize kernel -> kernel_w (output #2) + MSE vs gt_kernel
//   k_ebuild   : E[b] = kernel_w[b] (21x21) conv (r x r) -> 36x36 effective tap
//   k_conv     : lr_pred = hr_pred conv E, stride 4, via f32 WMMA 16x16x4 with
//                LDS Toeplitz B; image tile staged with async global->LDS
//   k_final    : combine partials -> loss[b] (output #1)
// ---------------------------------------------------------------------------

typedef __attribute__((ext_vector_type(2))) float v2f;
typedef __attribute__((ext_vector_type(8))) float v8f;
typedef __attribute__((ext_vector_type(2))) int   v2i;

#define B_  16
#define C_  3
#define H_  512
#define W_  512
#define HL  128
#define WL  128
#define KS  21
#define KK  441          // 21*21
#define SP  4096         // 64*64
#define NE  36           // effective kernel extent (21 + 16 - 1)
#define SCH 8            // spatial chunks in k_ksum

// workspace layout (float offsets); total 82720 floats = 330880 bytes
#define WS_HRPART 0          // [16][96]
#define WS_LRPART 1536       // [16][192]
#define WS_KLOSS  4608       // [16]
#define WS_KSUM   4624       // [16][8][448] partials per spatial chunk
#define WS_R      61968      // [16]
#define WS_E      61984      // [16][1296]

#if __has_builtin(__builtin_amdgcn_global_load_async_to_lds_b64)
#define HAVE_ASYNC 1
typedef __attribute__((address_space(1))) v2i* gptr_v2i;   // global v2i*
typedef __attribute__((address_space(3))) v2i* lptr_v2i;   // LDS v2i*
// global: AS1 pointers share representation with generic -> int round-trip ok
#define ASG(p) ((gptr_v2i)(uintptr_t)(p))
// LDS: needs a real addrspacecast (AS3 is a 32-bit LDS offset)
#define ASL(p) ((lptr_v2i)(v2i*)(p))
#endif

static __device__ __forceinline__ void wait_async0() {
#if __has_builtin(__builtin_amdgcn_s_wait_asynccnt)
  __builtin_amdgcn_s_wait_asynccnt(0);
#else
  asm volatile("s_wait_asynccnt 0x0" ::: "memory");
#endif
}

// -------------------------------------------------------------- r weights ---
__global__ void k_rweights(float* __restrict__ ws) {
  if (threadIdx.x == 0) {
    float w[16], s = 0.f;
    for (int j = 0; j < 16; ++j) {
      float x = fabsf((j - 7.5f) * 0.25f);       // Keys cubic, a = -0.5
      float v = (x <= 1.f) ? (( 1.5f * x - 2.5f) * x) * x + 1.f
                           : ((-0.5f * x + 2.5f) * x - 4.f) * x + 2.f;
      w[j] = v; s += v;
    }
    for (int j = 0; j < 16; ++j) ws[WS_R + j] = w[j] / s;
  }
}

// ----------------------------------------- kernel_pred GAP via WMMA f32 -----
// wave job = (b, spatial chunk sc, ktile): A[16x4] = 4 spatial vals per kernel
// row, B = ones => every column of C holds the partial row sum.
// Rows 441..447 are clamped duplicates of row 440 and are never read.
__global__ void __launch_bounds__(128) k_ksum(const float* __restrict__ kpred,
                                              float* __restrict__ ws) {
  int wave = threadIdx.x >> 5, lane = threadIdx.x & 31;
  int job = blockIdx.x * 4 + wave;          // 0..3583 == 16*8*28
  int b = job / (SCH * 28);
  int r2 = job % (SCH * 28);
  int sc = r2 / 28, kt = r2 % 28;
  int row  = kt * 16 + (lane & 15);
  int rowc = row < KK ? row : (KK - 1);     // branch-free clamp
  int koff = (lane >> 4) * 2;               // lanes 16..31 hold K={2,3}
  const float* src = kpred + ((size_t)b * KK + rowc) * SP + sc * (SP / SCH) + koff;
  v8f acc = {0.f,0.f,0.f,0.f,0.f,0.f,0.f,0.f};
  v2f ones; ones.x = 1.f; ones.y = 1.f;
  for (int s = 0; s < SP / SCH; s += 32) {  // 16 iters, 8 loads in flight
    v2f a[8];
#pragma unroll
    for (int u = 0; u < 8; ++u) a[u] = *(const v2f*)(src + s + 4 * u);
#pragma unroll
    for (int u = 0; u < 8; ++u)
      acc = __builtin_amdgcn_wmma_f32_16x16x4_f32(false, a[u], false, ones,
                                                  (short)0, acc, false, false);
  }
  if ((lane & 15) == 0) {                   // N==0 column carries row sums
    int mbase = kt * 16 + (lane >> 4) * 8;
    for (int v = 0; v < 8; ++v)
      ws[WS_KSUM + (b * SCH + sc) * 448 + mbase + v] = acc[v];
  }
}

// -------------------------------------------------- hr L1 partial sums ------
__global__ void k_hrl1(const float* __restrict__ p, const float* __restrict__ t,
                       float* __restrict__ ws) {
  __shared__ float red[256];
  int b = blockIdx.y, seg = blockIdx.x;                 // 96 segs * 8192 elems
  size_t base = (size_t)b * (C_ * H_ * W_) + (size_t)seg * 8192;
  const float4* p4 = (const float4*)(p + base);
  const float4* t4 = (const float4*)(t + base);
  float s = 0.f;
#pragma unroll
  for (int k = 0; k < 8; ++k) {
    float4 a = p4[threadIdx.x + k * 256];
    float4 c = t4[threadIdx.x + k * 256];
    s += fabsf(a.x - c.x) + fabsf(a.y - c.y) + fabsf(a.z - c.z) + fabsf(a.w - c.w);
  }
  red[threadIdx.x] = s; __syncthreads();
  for (int off = 128; off; off >>= 1) {
    if ((int)threadIdx.x < off) red[threadIdx.x] += red[threadIdx.x + off];
    __syncthreads();
  }
  if (threadIdx.x == 0) ws[WS_HRPART + b * 96 + seg] = red[0];
}

// ------------------------------- normalize kernel + kernel MSE partials -----
__global__ void k_knorm(const float* __restrict__ gt, float* __restrict__ ws,
                        float* __restrict__ out) {
  __shared__ float red[512];
  int b = blockIdx.x, t = threadIdx.x;
  float v = 0.f;
  if (t < KK)
    for (int sc = 0; sc < SCH; ++sc) v += ws[WS_KSUM + (b * SCH + sc) * 448 + t];
  red[t] = v; __syncthreads();
  for (int off = 256; off; off >>= 1) {
    if (t < off) red[t] += red[t + off];
    __syncthreads();
  }
  float total = red[0]; __syncthreads();
  float e = 0.f;
  if (t < KK) {
    float kw = v / total;                   // GAP / sum  (means cancel)
    out[16 + b * KK + t] = kw;              // output #2: kernel_w
    float d = kw - gt[b * KK + t];
    e = d * d;
  }
  red[t] = e; __syncthreads();
  for (int off = 256; off; off >>= 1) {
    if (t < off) red[t] += red[t + off];
    __syncthreads();
  }
  if (t == 0) ws[WS_KLOSS + b] = red[0];
}

// --------------------- effective 36x36 kernel: E = kw (x) (r outer r) -------
__global__ void k_ebuild(const float* __restrict__ out, float* __restrict__ ws) {
  __shared__ float kw[KK];
  __shared__ float r[16];
  int b = blockIdx.x;
  for (int i = threadIdx.x; i < KK; i += 256) kw[i] = out[16 + b * KK + i];
  if (threadIdx.x < 16) r[threadIdx.x] = ws[WS_R + threadIdx.x];
  __syncthreads();
  for (int idx = threadIdx.x; idx < NE * NE; idx += 256) {
    int dy = idx / NE, dx = idx % NE;
    int plo = max(0, dy - 15), phi = min(20, dy);
    int qlo = max(0, dx - 15), qhi = min(20, dx);
    float s = 0.f;
    for (int p = plo; p <= phi; ++p) {
      float rp = r[dy - p];
      for (int q = qlo; q <= qhi; ++q) s += kw[p * KS + q] * rp * r[dx - q];
    }
    ws[WS_E + b * (NE * NE) + idx] = s;
  }
}

// --------------------- fused blur+bicubic-down4 via WMMA f32 16x16x4 --------
// One wave computes a 16x16 tile of lr_pred for one (b,c).
//   C[m,n] = sum_i sum_k  A[m,k] * B_i[k,n]
//   A[m,k] = img[4*(y0+m)+i-16, xwin0+k]      (LDS image tile, 96x96)
//   B_i[k,n] = E[i, k - 4n]                   (LDS Toeplitz, zero-guarded)
#define IS 98    // image LDS row stride (even => 8B aligned; 8m bank spread)
#define ES 162   // E LDS row stride (64 left guard + 96 span + pad, even)
__global__ void __launch_bounds__(32) k_conv(const float* __restrict__ hr,
                                             const float* __restrict__ lrt,
                                             float* __restrict__ ws) {
  __shared__ float simg[96 * IS];   // 36.8 KB
  __shared__ float sE[NE * ES];     // 22.8 KB
  int lane = threadIdx.x;
  int xt = blockIdx.x, yt = blockIdx.y;          // 8 x 8 tiles of 16x16
  int bc = blockIdx.z;                           // 48 = 16*3
  int b = bc / C_, c = bc % C_;

  // E (zero-guarded): valid taps at [dy*ES + 64 + dx]
  for (int i = lane; i < NE * ES; i += 32) sE[i] = 0.f;
  const float* Eb = ws + WS_E + b * (NE * NE);
  for (int i = lane; i < NE * NE; i += 32)
    sE[(i / NE) * ES + 64 + (i % NE)] = Eb[i];

  // image tile: rows 64*yt-16 .. +95, cols 64*xt-16 .. +95, zero padded
  int gr0 = yt * 64 - 16, gc0 = xt * 64 - 16;
  const float* img = hr + (size_t)(b * C_ + c) * (H_ * W_);
  bool interior = (gr0 >= 0) && (gr0 + 96 <= H_) && (gc0 >= 0) && (gc0 + 96 <= W_);
#ifdef HAVE_ASYNC
  if (interior) {                                // block-uniform branch
    const float* gbase = img + (size_t)gr0 * W_ + gc0;
    for (int g = lane; g < 48 * 96; g += 32) {   // 4608 8-byte chunks
      int rr = g / 48, cc = (g % 48) * 2;
      __builtin_amdgcn_global_load_async_to_lds_b64(
          ASG(gbase + (size_t)rr * W_ + cc), ASL(&simg[rr * IS + cc]), 0, 0);
    }
    wait_async0();
  } else
#endif
  {
    for (int rr = lane; rr < 96; rr += 32) {
      int gr = gr0 + rr;
      float* drow = &simg[rr * IS];
      if (gr < 0 || gr >= H_) {
        for (int j = 0; j < 96; ++j) drow[j] = 0.f;
      } else {
        const float* srow = img + (size_t)gr * W_;
        for (int j = 0; j < 96; ++j) {
          int gc = gc0 + j;
          drow[j] = (gc >= 0 && gc < W_) ? srow[gc] : 0.f;
        }
      }
    }
  }
  __syncthreads();

  int mA  = lane & 15;              // A: row m; B: column n (same lane id)
  int kof = (lane >> 4) * 2;        // K sub-offset {0,1} vs {2,3}
  v8f acc = {0.f,0.f,0.f,0.f,0.f,0.f,0.f,0.f};
  for (int i = 0; i < NE; ++i) {
    const float* arow = &simg[(4 * mA + i) * IS + kof];
    const float* brow = &sE[i * ES + 64 + kof - 4 * mA];   // j = kb+kof-4n
#pragma unroll
    for (int kb = 0; kb < 96; kb += 4) {
      v2f a  = *(const v2f*)(arow + kb);
      v2f bb = *(const v2f*)(brow + kb);
      acc = __builtin_amdgcn_wmma_f32_16x16x4_f32(false, a, false, bb,
                                                  (short)0, acc, false, false);
    }
  }

  // fused |lr_pred - lr_target|, wave-local deterministic reduce
  int y0 = yt * 16, x0 = xt * 16;
  int N = lane & 15, Mb = (lane >> 4) * 8;
  const float* lt = lrt + (size_t)(b * C_ + c) * (HL * WL);
  float s = 0.f;
#pragma unroll
  for (int v = 0; v < 8; ++v) {
    int y = y0 + Mb + v, x = x0 + N;
    s += fabsf(acc[v] - lt[y * WL + x]);
  }
  __syncthreads();
  simg[lane] = s;
  __syncthreads();
  if (lane == 0) {
    float tot = 0.f;
    for (int j = 0; j < 32; ++j) tot += simg[j];
    ws[WS_LRPART + b * 192 + c * 64 + yt * 8 + xt] = tot;
  }
}

// ------------------------------------------------------------ final loss ----
__global__ void k_final(const float* __restrict__ ws, float* __restrict__ out) {
  int b = threadIdx.x;
  if (b < B_) {
    float hs = 0.f, ls = 0.f;
    for (int i = 0; i < 96; ++i)  hs += ws[WS_HRPART + b * 96 + i];
    for (int i = 0; i < 192; ++i) ls += ws[WS_LRPART + b * 192 + i];
    out[b] = hs * (1.f / (C_ * H_ * W_))
           + ls * (1.f / (C_ * HL * WL))
           + ws[WS_KLOSS + b] * (1.f / KK);
  }
}

extern "C" void kernel_launch(void* const* d_in, const int* in_sizes, int n_in,
                              void* d_out, int out_size, void* d_ws, size_t ws_size,
                              hipStream_t stream) {
  const float* hr_pred   = (const float*)d_in[0];
  const float* hr_target = (const float*)d_in[1];
  const float* lr_target = (const float*)d_in[2];
  const float* kpred     = (const float*)d_in[3];
  const float* gtk       = (const float*)d_in[4];
  // d_in[5..6] (segments) unused by the reference loss path; d_in[7] = iter.
  float* out = (float*)d_out;       // [0..16) loss, [16..7072) kernel_w
  float* ws  = (float*)d_ws;        // needs 330880 bytes

  k_rweights<<<1, 32, 0, stream>>>(ws);
  k_ksum   <<<896, 128, 0, stream>>>(kpred, ws);
  k_hrl1   <<<dim3(96, 16), 256, 0, stream>>>(hr_pred, hr_target, ws);
  k_knorm  <<<16, 512, 0, stream>>>(gtk, ws, out);
  k_ebuild <<<16, 256, 0, stream>>>(out, ws);
  k_conv   <<<dim3(8, 8, 48), 32, 0, stream>>>(hr_pred, lr_target, ws);
  k_final  <<<1, 32, 0, stream>>>(ws, out);
}